// KGHeteConv_22402549416607
// MI455X (gfx1250) — compile-verified
//
#include <hip/hip_runtime.h>
#include <hip/hip_bf16.h>

#define D 128

typedef __attribute__((ext_vector_type(2))) float v2f;
typedef __attribute__((ext_vector_type(8))) float v8f;

__device__ __forceinline__ float sigmoidf_(float v) {
    return 1.0f / (1.0f + __expf(-v));
}

// ---------------------------------------------------------------------------
// Node GEMM: out[n, j] = sum_k X[n,k] * W[j,k] + b0[j] (+ b1[j] if b1)
// mode 0: store raw pre-activation.
// mode 1 (self layer): store (sigmoid(acc) + 0.5) * X[n, j]
// One wave per 16-row tile; full 16x128 stripe via V_WMMA_F32_16X16X4_F32
// (f32 A/B, f32 acc -> exact f32 GEMM semantics).
// ---------------------------------------------------------------------------
__global__ void __launch_bounds__(128) gemm_xWt(
    const float* __restrict__ X, const float* __restrict__ W,
    const float* __restrict__ b0, const float* __restrict__ b1,
    float* __restrict__ out, int ntiles, int mode)
{
    const int lane = threadIdx.x & 31;
    const int wave = threadIdx.x >> 5;
    const int tile = blockIdx.x * 4 + wave;
    if (tile >= ntiles) return;            // wave-uniform: EXEC stays all-1s
    const int row0 = tile * 16;

    const int nn   = lane & 15;            // M (for A) / N (for B, C, D)
    const int koff = (lane >> 4) << 1;     // 0 or 2: K sub-offset for this half-wave

    v8f acc[8];
#pragma unroll
    for (int c = 0; c < 8; ++c) {
        int col = c * 16 + nn;
        float bv = b0[col];
        if (b1) bv += b1[col];
#pragma unroll
        for (int r = 0; r < 8; ++r) acc[c][r] = bv;
    }

    const float* xrow = X + (size_t)(row0 + nn) * D + koff;

    for (int k = 0; k < D; k += 4) {
        v2f a = *(const v2f*)(xrow + k);
#pragma unroll
        for (int c = 0; c < 8; ++c) {
            v2f bf = *(const v2f*)(W + (size_t)(c * 16 + nn) * D + koff + k);
            acc[c] = __builtin_amdgcn_wmma_f32_16x16x4_f32(
                false, a, false, bf, (short)0, acc[c], false, false);
        }
    }

    const int mhi = (lane >> 4) << 3;      // 0 or 8
#pragma unroll
    for (int c = 0; c < 8; ++c) {
        size_t base = (size_t)(row0 + mhi) * D + c * 16 + nn;
#pragma unroll
        for (int r = 0; r < 8; ++r) {
            size_t idx = base + (size_t)r * D;
            float v = acc[c][r];
            if (mode == 1) v = (sigmoidf_(v) + 0.5f) * X[idx];
            out[idx] = v;
        }
    }
}

// ---------------------------------------------------------------------------
// CSR build: histogram -> exclusive scan -> bucket fill
// ---------------------------------------------------------------------------
__global__ void zero_i32(int* __restrict__ p, int n)
{
    int i = blockIdx.x * blockDim.x + threadIdx.x;
    int stride = gridDim.x * blockDim.x;
    for (; i < n; i += stride) p[i] = 0;
}

__global__ void count_edges(const int* __restrict__ dst, int E, int* __restrict__ counts)
{
    int i = blockIdx.x * blockDim.x + threadIdx.x;
    int stride = gridDim.x * blockDim.x;
    for (; i < E; i += stride) atomicAdd(&counts[dst[i]], 1);
}

// per-256-block exclusive scan; writes block totals
__global__ void __launch_bounds__(256) scan_block(
    const int* __restrict__ in, int* __restrict__ outExcl,
    int* __restrict__ blockSums, int n)
{
    __shared__ int sh[256];
    const int tid = threadIdx.x;
    const int g = blockIdx.x * 256 + tid;
    int v = (g < n) ? in[g] : 0;
    sh[tid] = v;
    __syncthreads();
    for (int off = 1; off < 256; off <<= 1) {
        int t = (tid >= off) ? sh[tid - off] : 0;
        __syncthreads();
        sh[tid] += t;
        __syncthreads();
    }
    if (g < n) outExcl[g] = sh[tid] - v;            // exclusive
    if (tid == 255) blockSums[blockIdx.x] = sh[tid]; // block total
}

// single-block exclusive scan of block sums (nb <= 512; here max 391)
__global__ void __launch_bounds__(512) scan_sums(int* __restrict__ sums, int nb)
{
    __shared__ int sh[512];
    const int tid = threadIdx.x;
    int v = (tid < nb) ? sums[tid] : 0;
    sh[tid] = v;
    __syncthreads();
    for (int off = 1; off < 512; off <<= 1) {
        int t = (tid >= off) ? sh[tid - off] : 0;
        __syncthreads();
        sh[tid] += t;
        __syncthreads();
    }
    if (tid < nb) sums[tid] = sh[tid] - v;           // exclusive
}

__global__ void __launch_bounds__(256) scan_add(
    int* __restrict__ offs, const int* __restrict__ sums, int n)
{
    int g = blockIdx.x * 256 + threadIdx.x;
    if (g < n) offs[g] += sums[blockIdx.x];
}

__global__ void fill_eidx(const int* __restrict__ dst, int E,
                          const int* __restrict__ offs, int* __restrict__ cursor,
                          int* __restrict__ eidx)
{
    int i = blockIdx.x * blockDim.x + threadIdx.x;
    int stride = gridDim.x * blockDim.x;
    for (; i < E; i += stride) {
        int d = dst[i];
        int p = atomicAdd(&cursor[d], 1);
        eidx[offs[d] + p] = i;
    }
}

// ---------------------------------------------------------------------------
// Gather + fused finalize: one wave per destination node, lane owns 4 comps.
//   acc = sum_e (Ypre[src]+ef*Wme) * (sigmoid(Gpre[src]+ef*Wge) - 0.5)
//   out[node] = relu(out[node](self term) + acc + bias)
// No f32 atomics; Ypre/Gpre gathers are coalesced b128 loads (512B per row).
// ---------------------------------------------------------------------------
__global__ void __launch_bounds__(256) gather_finalize(
    const int* __restrict__ eidx, const int* __restrict__ offs,
    const int* __restrict__ deg,
    const int* __restrict__ src, const float* __restrict__ ef,
    const float* __restrict__ Ypre, const float* __restrict__ Gpre,
    const float* __restrict__ Wme, const float* __restrict__ Wge,
    const float* __restrict__ bias, float* __restrict__ out, int nnodes)
{
    const int lane = threadIdx.x & 31;
    const int node = (int)((blockIdx.x * blockDim.x + threadIdx.x) >> 5);
    if (node >= nnodes) return;
    const int c0 = lane * 4;

    const float4 wme = *(const float4*)(Wme + c0);
    const float4 wge = *(const float4*)(Wge + c0);
    float4 acc = make_float4(0.f, 0.f, 0.f, 0.f);

    const int beg = offs[node];
    const int end = beg + deg[node];
    for (int j = beg; j < end; ++j) {
        int   e = eidx[j];          // wave-uniform
        int   s = src[e];           // wave-uniform
        float f = ef[e];            // wave-uniform
        const size_t rb = (size_t)s * D + c0;
        float4 y = *(const float4*)(Ypre + rb);
        float4 g = *(const float4*)(Gpre + rb);
        acc.x += (y.x + f * wme.x) * (sigmoidf_(g.x + f * wge.x) - 0.5f);
        acc.y += (y.y + f * wme.y) * (sigmoidf_(g.y + f * wge.y) - 0.5f);
        acc.z += (y.z + f * wme.z) * (sigmoidf_(g.z + f * wge.z) - 0.5f);
        acc.w += (y.w + f * wme.w) * (sigmoidf_(g.w + f * wge.w) - 0.5f);
    }

    const float4 bv = *(const float4*)(bias + c0);
    const size_t idx = (size_t)node * D + c0;
    float4 o = *(const float4*)(out + idx);
    o.x = fmaxf(o.x + acc.x + bv.x, 0.f);
    o.y = fmaxf(o.y + acc.y + bv.y, 0.f);
    o.z = fmaxf(o.z + acc.z + bv.z, 0.f);
    o.w = fmaxf(o.w + acc.w + bv.w, 0.f);
    *(float4*)(out + idx) = o;
}

// ---------------------------------------------------------------------------
extern "C" void kernel_launch(void* const* d_in, const int* in_sizes, int n_in,
                              void* d_out, int out_size, void* d_ws, size_t ws_size,
                              hipStream_t stream)
{
    const float* x_data  = (const float*)d_in[0];
    const float* x_task  = (const float*)d_in[1];
    const int*   src_dt  = (const int*)  d_in[2];
    const int*   dst_dt  = (const int*)  d_in[3];
    const float* ef_dt   = (const float*)d_in[4];
    const int*   src_td  = (const int*)  d_in[5];
    const int*   dst_td  = (const int*)  d_in[6];
    const float* ef_td   = (const float*)d_in[7];
    const float* Wmn_dt  = (const float*)d_in[8];
    const float* bmn_dt  = (const float*)d_in[9];
    const float* Wme_dt  = (const float*)d_in[10];
    const float* bme_dt  = (const float*)d_in[11];
    const float* Wgn_dt  = (const float*)d_in[12];
    const float* bgn_dt  = (const float*)d_in[13];
    const float* Wge_dt  = (const float*)d_in[14];
    const float* bge_dt  = (const float*)d_in[15];
    const float* bias_dt = (const float*)d_in[16];
    const float* Wmn_td  = (const float*)d_in[17];
    const float* bmn_td  = (const float*)d_in[18];
    const float* Wme_td  = (const float*)d_in[19];
    const float* bme_td  = (const float*)d_in[20];
    const float* Wgn_td  = (const float*)d_in[21];
    const float* bgn_td  = (const float*)d_in[22];
    const float* Wge_td  = (const float*)d_in[23];
    const float* bge_td  = (const float*)d_in[24];
    const float* bias_td = (const float*)d_in[25];
    const float* Ws_data = (const float*)d_in[26];
    const float* bs_data = (const float*)d_in[27];
    const float* Ws_task = (const float*)d_in[28];
    const float* bs_task = (const float*)d_in[29];

    const int Ndata = in_sizes[0] / D;   // 100000
    const int Ntask = in_sizes[1] / D;   // 20000
    const int E     = in_sizes[2];       // 1000000

    const size_t szD = (size_t)Ndata * D;
    const size_t szT = (size_t)Ntask * D;

    // workspace layout (4-byte units)
    float* ws        = (float*)d_ws;
    float* Ypre_dt   = ws;                               // [Ndata, D]
    float* Gpre_dt   = Ypre_dt + szD;                    // [Ndata, D]
    float* Ypre_td   = Gpre_dt + szD;                    // [Ntask, D]
    float* Gpre_td   = Ypre_td + szT;                    // [Ntask, D]
    int*   cnt_task  = (int*)(Gpre_td + szT);            // [Ntask]
    int*   cnt_data  = cnt_task + Ntask;                 // [Ndata]  (contiguous w/ cnt_task)
    int*   offs_task = cnt_data + Ndata;                 // [Ntask]
    int*   offs_data = offs_task + Ntask;                // [Ndata]
    int*   eidx_dt   = offs_data + Ndata;                // [E]
    int*   eidx_td   = eidx_dt + E;                      // [E]
    int*   bsum      = eidx_td + E;                      // [512] scratch (reused serially)

    float* out_data = (float*)d_out;          // rows [0, Ndata)
    float* out_task = out_data + szD;         // rows [Ndata, Ndata+Ntask)

    const int nbT = (Ntask + 255) / 256;      // 79  (<= 512 required)
    const int nbD = (Ndata + 255) / 256;      // 391 (<= 512 required)

    // ---- 1) CSR build for both edge directions ----
    zero_i32<<<1024, 256, 0, stream>>>(cnt_task, Ntask + Ndata);
    count_edges<<<1024, 256, 0, stream>>>(dst_dt, E, cnt_task);
    count_edges<<<1024, 256, 0, stream>>>(dst_td, E, cnt_data);

    scan_block<<<nbT, 256, 0, stream>>>(cnt_task, offs_task, bsum, Ntask);
    scan_sums<<<1, 512, 0, stream>>>(bsum, nbT);
    scan_add<<<nbT, 256, 0, stream>>>(offs_task, bsum, Ntask);

    scan_block<<<nbD, 256, 0, stream>>>(cnt_data, offs_data, bsum, Ndata);
    scan_sums<<<1, 512, 0, stream>>>(bsum, nbD);
    scan_add<<<nbD, 256, 0, stream>>>(offs_data, bsum, Ndata);

    zero_i32<<<1024, 256, 0, stream>>>(cnt_task, Ntask + Ndata);   // cursors
    fill_eidx<<<1024, 256, 0, stream>>>(dst_dt, E, offs_task, cnt_task, eidx_dt);
    fill_eidx<<<1024, 256, 0, stream>>>(dst_td, E, offs_data, cnt_data, eidx_td);
    // after fill, cnt_* hold node degrees again

    // ---- 2) per-node GEMMs (f32 WMMA); msg/gate biases folded ----
    const int tilesD = Ndata / 16, tilesT = Ntask / 16;
    gemm_xWt<<<(tilesD + 3) / 4, 128, 0, stream>>>(x_data, Wmn_dt, bmn_dt, bme_dt, Ypre_dt, tilesD, 0);
    gemm_xWt<<<(tilesD + 3) / 4, 128, 0, stream>>>(x_data, Wgn_dt, bgn_dt, bge_dt, Gpre_dt, tilesD, 0);
    gemm_xWt<<<(tilesT + 3) / 4, 128, 0, stream>>>(x_task, Wmn_td, bmn_td, bme_td, Ypre_td, tilesT, 0);
    gemm_xWt<<<(tilesT + 3) / 4, 128, 0, stream>>>(x_task, Wgn_td, bgn_td, bge_td, Gpre_td, tilesT, 0);
    // self layers written straight into d_out
    gemm_xWt<<<(tilesD + 3) / 4, 128, 0, stream>>>(x_data, Ws_data, bs_data, nullptr, out_data, tilesD, 1);
    gemm_xWt<<<(tilesT + 3) / 4, 128, 0, stream>>>(x_task, Ws_task, bs_task, nullptr, out_task, tilesT, 1);

    // ---- 3) gather + fused finalize (atomic-free segment sum) ----
    gather_finalize<<<(Ntask + 7) / 8, 256, 0, stream>>>(
        eidx_dt, offs_task, cnt_task, src_dt, ef_dt,
        Ypre_dt, Gpre_dt, Wme_dt, Wge_dt, bias_dt, out_task, Ntask);
    gather_finalize<<<(Ndata + 7) / 8, 256, 0, stream>>>(
        eidx_td, offs_data, cnt_data, src_td, ef_td,
        Ypre_td, Gpre_td, Wme_td, Wge_td, bias_td, out_data, Ndata);
}